// PointNet2ClsMsg_61967788147273
// MI455X (gfx1250) — compile-verified
//
#include <hip/hip_runtime.h>

typedef __attribute__((ext_vector_type(2))) float v2f;
typedef __attribute__((ext_vector_type(8))) float v8f;

#define FAST_MAXK 768   // max padded K handled by the LDS-staged fast GEMM (sa3 layer1 = 644)

#define AS1 __attribute__((address_space(1)))
#define AS3 __attribute__((address_space(3)))

// ----------------------------------------------------------------------------
// Elementwise / data-movement kernels
// ----------------------------------------------------------------------------

__global__ void k_transpose_pc(const float* __restrict__ pc, float* __restrict__ xyz,
                               int B, int N) {
  int i = blockIdx.x * blockDim.x + threadIdx.x;
  if (i >= B * N) return;
  int b = i / N, n = i - b * N;
#pragma unroll
  for (int c = 0; c < 3; ++c)
    xyz[((size_t)b * N + n) * 3 + c] = pc[((size_t)b * 3 + c) * N + n];
}

// Farthest point sampling: one block per batch, iterative with block argmax.
__global__ void k_fps(const float* __restrict__ xyz, float* __restrict__ dist,
                      int* __restrict__ out_idx, int N, int S) {
  int b = blockIdx.x;
  int t = threadIdx.x;
  const int T = 256;
  const float* x = xyz + (size_t)b * N * 3;
  float* d = dist + (size_t)b * N;
  __shared__ float s_val[256];
  __shared__ int s_idx[256];
  __shared__ int s_far;
  if (t == 0) s_far = 0;
  for (int n = t; n < N; n += T) d[n] = 1e10f;
  __syncthreads();
  for (int it = 0; it < S; ++it) {
    int far = s_far;
    if (t == 0) out_idx[b * S + it] = far;   // record far BEFORE update (ref scan)
    float cx = x[far * 3 + 0], cy = x[far * 3 + 1], cz = x[far * 3 + 2];
    float best = -1.0f;
    int bi = 0;
    for (int n = t; n < N; n += T) {
      float dx = x[n * 3 + 0] - cx, dy = x[n * 3 + 1] - cy, dz = x[n * 3 + 2] - cz;
      float dd = dx * dx + dy * dy + dz * dz;
      float dn = fminf(d[n], dd);
      d[n] = dn;
      if (dn > best) { best = dn; bi = n; }
    }
    s_val[t] = best; s_idx[t] = bi;
    __syncthreads();
    for (int off = T >> 1; off > 0; off >>= 1) {
      if (t < off) {
        float v2 = s_val[t + off]; int i2 = s_idx[t + off];
        if (v2 > s_val[t] || (v2 == s_val[t] && i2 < s_idx[t])) { s_val[t] = v2; s_idx[t] = i2; }
      }
      __syncthreads();
    }
    if (t == 0) s_far = s_idx[0];
    __syncthreads();
  }
}

// out[i, :] = pts[b, idx[i], :]   (i over B*S)
__global__ void k_gather(const float* __restrict__ pts, const int* __restrict__ idx,
                         float* __restrict__ out, int N, int S, int C, int total) {
  int i = blockIdx.x * blockDim.x + threadIdx.x;
  if (i >= total) return;
  int b = i / S;
  int gi = idx[i];
  for (int c = 0; c < C; ++c)
    out[(size_t)i * C + c] = pts[((size_t)b * N + gi) * C + c];
}

// Ball query: first K in-range indices in ascending order, pad with first hit.
__global__ void k_ballquery(const float* __restrict__ xyz, const float* __restrict__ nxyz,
                            int* __restrict__ gidx, int N, int S, int K, float r2, int total) {
  int i = blockIdx.x * blockDim.x + threadIdx.x;
  if (i >= total) return;  // total = B*S
  int b = i / S;
  float qx = nxyz[(size_t)i * 3 + 0], qy = nxyz[(size_t)i * 3 + 1], qz = nxyz[(size_t)i * 3 + 2];
  const float* x = xyz + (size_t)b * N * 3;
  int* g = gidx + (size_t)i * K;
  int cnt = 0, first = 0;
  bool hf = false;
  for (int n = 0; n < N && cnt < K; ++n) {
    float dx = x[n * 3 + 0] - qx, dy = x[n * 3 + 1] - qy, dz = x[n * 3 + 2] - qz;
    float d2 = dx * dx + dy * dy + dz * dz;
    if (d2 <= r2) {
      if (!hf) { first = n; hf = true; }
      g[cnt++] = n;
    }
  }
  for (; cnt < K; ++cnt) g[cnt] = first;
}

// Grouped features, K-padded: row ((b*S+s)*K+k) =
//   [ points[b,gidx,:Cpts], xyz[gidx]-new_xyz, 0...0 (to Kpad) ]
__global__ void k_group(const float* __restrict__ xyz, const float* __restrict__ nxyz,
                        const float* __restrict__ pts, const int* __restrict__ gidx,
                        float* __restrict__ A, int N, int S, int K, int Cpts, int Kpad,
                        int total) {
  int i = blockIdx.x * blockDim.x + threadIdx.x;
  if (i >= total) return;  // total = B*S*K
  int k = i % K;
  int bs = i / K;
  int b = bs / S;
  int gi = gidx[(size_t)bs * K + k];
  int Cin = Cpts + 3;
  float* row = A + (size_t)i * Kpad;
  if (pts) {
    const float* pp = pts + ((size_t)b * N + gi) * Cpts;
    for (int c = 0; c < Cpts; ++c) row[c] = pp[c];
  }
#pragma unroll
  for (int c = 0; c < 3; ++c)
    row[Cpts + c] = xyz[((size_t)b * N + gi) * 3 + c] - nxyz[(size_t)bs * 3 + c];
  for (int c = Cin; c < Kpad; ++c) row[c] = 0.0f;
}

// sa3 group-all input (K-padded): row i(=b*S+n) = [ xyz[i,:3], pts[i,:Cpts], 0pad ]
__global__ void k_concat_xyz_pts(const float* __restrict__ xyz, const float* __restrict__ pts,
                                 float* __restrict__ A, int rows, int Cpts, int Kpad) {
  int i = blockIdx.x * blockDim.x + threadIdx.x;
  if (i >= rows) return;
  float* row = A + (size_t)i * Kpad;
  row[0] = xyz[(size_t)i * 3 + 0];
  row[1] = xyz[(size_t)i * 3 + 1];
  row[2] = xyz[(size_t)i * 3 + 2];
  for (int c = 0; c < Cpts; ++c) row[3 + c] = pts[(size_t)i * Cpts + c];
  for (int c = Cpts + 3; c < Kpad; ++c) row[c] = 0.0f;
}

// ----------------------------------------------------------------------------
// Fast fp32 WMMA GEMM (SA stages): C[M,N] = A[M,Kpad] @ W[K,N] + bias.
// Preconditions: M % 64 == 0, N % 16 == 0, Kpad % 4 == 0, Kpad <= FAST_MAXK,
// A rows zero-padded from K..Kpad. Block = 128 threads = 4 waves -> 64x16 tile.
// The 16-column W tile is staged TRANSPOSED into LDS. Staging uses the CDNA5
// async global->LDS path (per-lane LDS destination addresses do the transpose
// with no VGPR round trip), tracked by ASYNCcnt; falls back to a VGPR-bounce
// copy if the builtin is absent. Inner loop: global_load_b64 (A-frag) +
// ds_load_b64 (B-frag) + v_wmma_f32_16x16x4_f32.
// ----------------------------------------------------------------------------
__global__ void k_gemm_fast(const float* __restrict__ A, const float* __restrict__ W,
                            const float* __restrict__ bias, float* __restrict__ C,
                            int K, int Kpad, int N) {
  __shared__ float ldsW[16 * FAST_MAXK];   // [n][k], row stride Kpad

  int col0 = blockIdx.x * 16;
  int row0 = blockIdx.y * 64 + (threadIdx.x >> 5) * 16;
  int lane = threadIdx.x & 31;
  int lo = lane & 15;
  int half = lane >> 4;

  // Stage W[:,col0:col0+16] transposed, zero-padding k in [K, Kpad).
#if __has_builtin(__builtin_amdgcn_global_load_async_to_lds_b32)
  for (int idx = threadIdx.x; idx < 16 * Kpad; idx += 128) {
    int k = idx >> 4, n = idx & 15;
    if (k < K) {
      __builtin_amdgcn_global_load_async_to_lds_b32(
          (AS1 int*)(void*)(W + (size_t)k * N + col0 + n),
          (AS3 int*)(void*)&ldsW[n * Kpad + k], 0, 0);
    } else {
      ldsW[n * Kpad + k] = 0.0f;
    }
  }
#if __has_builtin(__builtin_amdgcn_s_wait_asynccnt)
  __builtin_amdgcn_s_wait_asynccnt(0);
#else
  asm volatile("s_wait_asynccnt 0" ::: "memory");
#endif
#else
  for (int idx = threadIdx.x; idx < 16 * Kpad; idx += 128) {
    int k = idx >> 4, n = idx & 15;
    ldsW[n * Kpad + k] = (k < K) ? W[(size_t)k * N + col0 + n] : 0.0f;
  }
#endif
  __syncthreads();

  const float* ap = A + (size_t)(row0 + lo) * Kpad + 2 * half;
  const float* wp = &ldsW[lo * Kpad + 2 * half];

  v8f acc = {};
  for (int ka = 0; ka < Kpad; ka += 4) {
    __builtin_prefetch(ap + ka + 128, 0, 0);      // global_prefetch_b8, ~2 lines ahead
    v2f av = *(const v2f*)(ap + ka);              // A-frag: K={ka+2h, ka+2h+1}
    v2f bv = *(const v2f*)(wp + ka);              // B-frag: same K pair, col = lo
    acc = __builtin_amdgcn_wmma_f32_16x16x4_f32(false, av, false, bv,
                                                (short)0, acc, false, false);
  }

  int n = col0 + lo;
  float bb = bias[n];
#pragma unroll
  for (int i = 0; i < 8; ++i)
    C[(size_t)(row0 + i + half * 8) * N + n] = acc[i] + bb;
}

// Guarded WMMA GEMM for the tiny FC head (M=8, N=40 tail, K up to 1024).
__global__ void k_gemm_bias_wmma(const float* __restrict__ A, const float* __restrict__ W,
                                 const float* __restrict__ bias, float* __restrict__ C,
                                 int M, int K, int N) {
  int lane = threadIdx.x & 31;
  int wv = threadIdx.x >> 5;
  int row0 = blockIdx.y * 64 + wv * 16;
  int col0 = blockIdx.x * 16;
  int lo = lane & 15;
  int half = lane >> 4;
  int arow = row0 + lo;
  bool rok = arow < M;
  const float* arp = A + (size_t)(rok ? arow : 0) * K;
  int bcol = col0 + lo;
  bool cok = bcol < N;

  v8f acc = {};
  for (int kk = 0; kk < K; kk += 4) {
    int ka = kk + half * 2;
    v2f a, bf;
    a.x  = (rok && ka     < K) ? arp[ka]     : 0.0f;
    a.y  = (rok && ka + 1 < K) ? arp[ka + 1] : 0.0f;
    bf.x = (cok && ka     < K) ? W[(size_t)ka * N + bcol]       : 0.0f;
    bf.y = (cok && ka + 1 < K) ? W[(size_t)(ka + 1) * N + bcol] : 0.0f;
    acc = __builtin_amdgcn_wmma_f32_16x16x4_f32(false, a, false, bf,
                                                (short)0, acc, false, false);
  }
  float bb = cok ? bias[bcol] : 0.0f;
#pragma unroll
  for (int i = 0; i < 8; ++i) {
    int m = row0 + i + half * 8;
    if (m < M && cok) C[(size_t)m * N + bcol] = acc[i] + bb;
  }
}

// Per-channel BN stats over M rows -> scale = g*rsqrt(var+eps), shift = beta - mean*scale
__global__ void k_bn_stats(const float* __restrict__ X, const float* __restrict__ g,
                           const float* __restrict__ beta, float* __restrict__ scale,
                           float* __restrict__ shift, int M, int C) {
  int c = blockIdx.x;
  int t = threadIdx.x;
  __shared__ float s1[256], s2[256];
  float a = 0.0f, b = 0.0f;
  for (int m = t; m < M; m += 256) {
    float v = X[(size_t)m * C + c];
    a += v; b += v * v;
  }
  s1[t] = a; s2[t] = b;
  __syncthreads();
  for (int off = 128; off > 0; off >>= 1) {
    if (t < off) { s1[t] += s1[t + off]; s2[t] += s2[t + off]; }
    __syncthreads();
  }
  if (t == 0) {
    float mean = s1[0] / (float)M;
    float var = s2[0] / (float)M - mean * mean;
    float sc = g[c] * rsqrtf(var + 1e-5f);
    scale[c] = sc;
    shift[c] = beta[c] - mean * sc;
  }
}

__global__ void k_bn_relu(float* __restrict__ X, const float* __restrict__ scale,
                          const float* __restrict__ shift, size_t total, int C) {
  size_t i = (size_t)blockIdx.x * blockDim.x + threadIdx.x;
  if (i >= total) return;
  int c = (int)(i % (size_t)C);
  float v = X[i] * scale[c] + shift[c];
  X[i] = v > 0.0f ? v : 0.0f;
}

// Max over neighborhood K: O[r, coff+c] = max_k X[(r*K+k)*C + c]
__global__ void k_maxpool(const float* __restrict__ X, float* __restrict__ O,
                          int rows, int K, int C, int Ctot, int coff) {
  int i = blockIdx.x * blockDim.x + threadIdx.x;
  if (i >= rows * C) return;
  int r = i / C, c = i - r * C;
  const float* p = X + (size_t)r * K * C + c;
  float m = p[0];
  for (int k = 1; k < K; ++k) m = fmaxf(m, p[(size_t)k * C]);
  O[(size_t)r * Ctot + coff + c] = m;
}

// ----------------------------------------------------------------------------
// Host orchestration
// ----------------------------------------------------------------------------
extern "C" void kernel_launch(void* const* d_in, const int* in_sizes, int n_in,
                              void* d_out, int out_size, void* d_ws, size_t ws_size,
                              hipStream_t stream) {
  (void)in_sizes; (void)n_in; (void)out_size; (void)ws_size;
  const int B = 8, N = 4096;

  int p = 0;
  const float* pc = (const float*)d_in[p++];

  struct LayerP { const float *W, *b, *g, *be; int cin, cout; };
  auto take = [&](int cin, int cout) -> LayerP {
    LayerP L;
    L.W = (const float*)d_in[p++]; L.b = (const float*)d_in[p++];
    L.g = (const float*)d_in[p++]; L.be = (const float*)d_in[p++];
    L.cin = cin; L.cout = cout;
    return L;
  };

  LayerP sa1[3][3], sa2[3][3], sa3[3], fc[3];
  const int w1[3][3] = {{32, 32, 64}, {64, 64, 128}, {64, 96, 128}};
  for (int br = 0; br < 3; ++br) { int c = 3;   for (int l = 0; l < 3; ++l) { sa1[br][l] = take(c, w1[br][l]); c = w1[br][l]; } }
  const int w2[3][3] = {{64, 64, 128}, {128, 128, 256}, {128, 128, 256}};
  for (int br = 0; br < 3; ++br) { int c = 323; for (int l = 0; l < 3; ++l) { sa2[br][l] = take(c, w2[br][l]); c = w2[br][l]; } }
  { const int w3[3] = {256, 512, 1024}; int c = 643; for (int l = 0; l < 3; ++l) { sa3[l] = take(c, w3[l]); c = w3[l]; } }
  fc[0] = take(1024, 512); fc[1] = take(512, 256); fc[2] = take(256, 40);

  // ---- workspace carve-up ----
  char* wsp = (char*)d_ws;
  auto allocf = [&](size_t n) -> float* { float* r = (float*)wsp; wsp += n * sizeof(float); return r; };
  auto alloci = [&](size_t n) -> int*   { int*   r = (int*)wsp;   wsp += n * sizeof(int);   return r; };
  float* xyz1  = allocf((size_t)B * N * 3);
  float* fdist = allocf((size_t)B * N);
  int*   fidx  = alloci((size_t)B * 512);
  int*   gidx  = alloci((size_t)B * 512 * 128);
  float* l1xyz = allocf((size_t)B * 512 * 3);
  float* l2xyz = allocf((size_t)B * 128 * 3);
  float* l1pts = allocf((size_t)B * 512 * 320);
  float* l2pts = allocf((size_t)B * 128 * 640);
  float* scale = allocf(1024);
  float* shift = allocf(1024);
  float* fcb1  = allocf((size_t)B * 512);
  float* fcb2  = allocf((size_t)B * 256);
  float* bufA  = allocf((size_t)B * 512 * 128 * 128);  // 67.1M floats (worst case sa1 br3)
  float* bufB  = allocf((size_t)B * 512 * 128 * 128);

  float* logits = (float*)d_out;        // 8 x 40
  float* l3pts  = (float*)d_out + 320;  // 8 x 1024

  auto g1d = [](int total) { return dim3((unsigned)((total + 255) / 256)); };

  // GEMM dispatch: LDS-staged fast kernel when shapes allow, guarded otherwise.
  auto gemmL = [&](const float* Ain, int Ktrue, int Kpad, const LayerP& Lp, float* Cout, int M) {
    if ((M % 64) == 0 && (Lp.cout % 16) == 0 && Kpad <= FAST_MAXK) {
      dim3 g((unsigned)(Lp.cout / 16), (unsigned)(M / 64));
      k_gemm_fast<<<g, 128, 0, stream>>>(Ain, Lp.W, Lp.b, Cout, Ktrue, Kpad, Lp.cout);
    } else {
      dim3 g((unsigned)((Lp.cout + 15) / 16), (unsigned)((M + 63) / 64));
      k_gemm_bias_wmma<<<g, 128, 0, stream>>>(Ain, Lp.W, Lp.b, Cout, M, Ktrue, Lp.cout);
    }
  };
  auto bnrelu = [&](float* X, const LayerP& Lp, int M) {
    k_bn_stats<<<dim3((unsigned)Lp.cout), 256, 0, stream>>>(X, Lp.g, Lp.be, scale, shift, M, Lp.cout);
    size_t tot = (size_t)M * Lp.cout;
    k_bn_relu<<<dim3((unsigned)((tot + 255) / 256)), 256, 0, stream>>>(X, scale, shift, tot, Lp.cout);
  };
  auto run_branch = [&](const float* srcxyz, int Nsrc, const float* nxyz, int S,
                        const float* pts, int Cpts, float radius, int K,
                        const LayerP* layers, float* outfeat, int Ctot, int coff) {
    int totq = B * S;
    k_ballquery<<<g1d(totq), 256, 0, stream>>>(srcxyz, nxyz, gidx, Nsrc, S, K, radius * radius, totq);
    int totg = B * S * K;
    int Cin = Cpts + 3;
    int Kp = (Cin + 3) & ~3;
    k_group<<<g1d(totg), 256, 0, stream>>>(srcxyz, nxyz, pts, gidx, bufA, Nsrc, S, K, Cpts, Kp, totg);
    float* cur = bufA; float* nxt = bufB;
    for (int l = 0; l < 3; ++l) {
      int kt = layers[l].cin;
      int kp = (l == 0) ? Kp : kt;   // later layers: cin already a multiple of 4
      gemmL(cur, kt, kp, layers[l], nxt, totg);
      bnrelu(nxt, layers[l], totg);
      float* t = cur; cur = nxt; nxt = t;
    }
    int C = layers[2].cout;
    k_maxpool<<<g1d(B * S * C), 256, 0, stream>>>(cur, outfeat, B * S, K, C, Ctot, coff);
  };

  // ---- forward pass ----
  k_transpose_pc<<<g1d(B * N), 256, 0, stream>>>(pc, xyz1, B, N);

  // SA1 (npoint=512)
  k_fps<<<dim3(B), 256, 0, stream>>>(xyz1, fdist, fidx, N, 512);
  k_gather<<<g1d(B * 512), 256, 0, stream>>>(xyz1, fidx, l1xyz, N, 512, 3, B * 512);
  {
    const float r1[3] = {0.1f, 0.2f, 0.4f};
    const int ns1[3] = {16, 32, 128};
    int coff = 0;
    for (int br = 0; br < 3; ++br) {
      run_branch(xyz1, N, l1xyz, 512, nullptr, 0, r1[br], ns1[br], sa1[br], l1pts, 320, coff);
      coff += w1[br][2];
    }
  }

  // SA2 (npoint=128)
  k_fps<<<dim3(B), 256, 0, stream>>>(l1xyz, fdist, fidx, 512, 128);
  k_gather<<<g1d(B * 128), 256, 0, stream>>>(l1xyz, fidx, l2xyz, 512, 128, 3, B * 128);
  {
    const float r2[3] = {0.2f, 0.4f, 0.8f};
    const int ns2[3] = {32, 64, 128};
    int coff = 0;
    for (int br = 0; br < 3; ++br) {
      run_branch(l1xyz, 512, l2xyz, 128, l1pts, 320, r2[br], ns2[br], sa2[br], l2pts, 640, coff);
      coff += w2[br][2];
    }
  }

  // SA3 (group all, K=128 neighborhood, rows = B*128, Cin=643 -> Kpad=644)
  {
    int Kp = 644;
    k_concat_xyz_pts<<<g1d(B * 128), 256, 0, stream>>>(l2xyz, l2pts, bufA, B * 128, 640, Kp);
    float* cur = bufA; float* nxt = bufB;
    int M = B * 128;
    for (int l = 0; l < 3; ++l) {
      int kt = sa3[l].cin;
      int kp = (l == 0) ? Kp : kt;
      gemmL(cur, kt, kp, sa3[l], nxt, M);
      bnrelu(nxt, sa3[l], M);
      float* t = cur; cur = nxt; nxt = t;
    }
    k_maxpool<<<g1d(B * 1024), 256, 0, stream>>>(cur, l3pts, B, 128, 1024, 1024, 0);
  }

  // FC head (tiny: guarded WMMA path)
  gemmL(l3pts, 1024, 1024, fc[0], fcb1, B); bnrelu(fcb1, fc[0], B);
  gemmL(fcb1, 512, 512, fc[1], fcb2, B);    bnrelu(fcb2, fc[1], B);
  gemmL(fcb2, 256, 256, fc[2], logits, B);  // no BN/ReLU on fc3
}